// PointConvNet3_50397146251471
// MI455X (gfx1250) — compile-verified
//
#include <hip/hip_runtime.h>

typedef __attribute__((ext_vector_type(2))) float v2f;
typedef __attribute__((ext_vector_type(8))) float v8f;

constexpr int B = 8, N = 4096, F = 8, H = 64, O = 128, K = 40;
constexpr int KE = 48;   // edge rows padded: 41 (K + self) -> 48 = 3 x 16
constexpr int FIN = 12;  // input features padded: 11 -> 12 = 3 x 4
constexpr int WPB = 2;   // waves (nodes) per block in the WMMA kernel

// --------------------------------------------------------------------------
// Kernel 1: exact KNN (loop=False) per batch. One block = 256 target points
// of one batch; all 4096 positions of the batch live in LDS (48 KB). Each
// thread keeps a sorted top-40 (insertion on strict improvement => stable
// tie-breaking by lower index, matching top_k). Also emits edge_index.
// --------------------------------------------------------------------------
__global__ __launch_bounds__(256) void knn_topk_kernel(
    const float* __restrict__ pos, int* __restrict__ idx_ws,
    float* __restrict__ out_full) {
  __shared__ float px[N], py[N], pz[N];
  const int tid = threadIdx.x;
  const int b = blockIdx.x / (N / 256);
  const int tile = blockIdx.x % (N / 256);
  for (int j = tid; j < N; j += 256) {
    const float* p = pos + ((size_t)b * N + j) * 3;
    px[j] = p[0]; py[j] = p[1]; pz[j] = p[2];
  }
  __syncthreads();

  const int i = tile * 256 + tid;
  const float xi = px[i], yi = py[i], zi = pz[i];
  float bd[K]; int bi[K];
  for (int k = 0; k < K; ++k) { bd[k] = 3.4e38f; bi[k] = 0; }
  for (int j = 0; j < N; ++j) {
    float dx = px[j] - xi, dy = py[j] - yi, dz = pz[j] - zi;
    float d = dx * dx + dy * dy + dz * dz;
    if (j == i) continue;  // loop=False (diag = inf)
    if (d < bd[K - 1]) {
      int p = K - 1;
      while (p > 0 && bd[p - 1] > d) { bd[p] = bd[p - 1]; bi[p] = bi[p - 1]; --p; }
      bd[p] = d; bi[p] = j;
    }
  }

  const size_t base = ((size_t)b * N + i) * K;
  const size_t eo = (size_t)B * N * O;       // out part comes first in d_out
  const size_t half = (size_t)B * N * K;     // edge_index row stride
  for (int k = 0; k < K; ++k) {
    idx_ws[base + k] = bi[k];
    out_full[eo + base + k] = (float)(bi[k] + b * N);         // src
    out_full[eo + half + base + k] = (float)(i + b * N);      // tgt
  }
}

// --------------------------------------------------------------------------
// Kernel 2: per-node edge MLP with fp32 WMMA (V_WMMA_F32_16X16X4_F32) and
// max aggregation. One wave per node, 2 waves per block. W2 staged once per
// block into LDS in pair-transposed layout so every B operand is one
// ds_load_b64; A operands are aligned v2f LDS loads.
// --------------------------------------------------------------------------
__global__ __launch_bounds__(32 * WPB) void pointconv_wmma_kernel(
    const float* __restrict__ x, const float* __restrict__ pos,
    const float* __restrict__ W1, const float* __restrict__ b1,
    const float* __restrict__ W2, const float* __restrict__ b2,
    const int* __restrict__ idx_ws, float* __restrict__ out) {
  __shared__ v2f sW2p[(H / 2) * O];     // [p][n] = {W2[2p][n], W2[2p+1][n]}
  __shared__ float smsg[WPB][KE][FIN];  // gathered [x_j, pos_j - pos_i, 0]
  __shared__ float sh[WPB][KE][H];      // hidden activations (C->A relayout)

  const int lane = threadIdx.x & 31;
  const int wid = threadIdx.x >> 5;
  const long node = (long)blockIdx.x * WPB + wid;
  const int b = (int)(node / N);
  const int i = (int)(node % N);
  const size_t nbase = (size_t)b * N + i;

  const int nloc = lane & 15;            // N/M position within a 16-tile
  const int k0 = (lane < 16) ? 0 : 2;    // K pair held by this lane half

  // Stage W2 pair-transposed into LDS (once per block)
  for (int t = threadIdx.x; t < (H / 2) * O; t += 32 * WPB) {
    int p = t >> 7, n = t & (O - 1);
    v2f w;
    w.x = W2[(size_t)(2 * p) * O + n];
    w.y = W2[(size_t)(2 * p + 1) * O + n];
    sW2p[t] = w;
  }

  // W1 B-operand tiles in registers: [k-step 0..2][h-tile 0..3]
  v2f w1b[3][4];
  for (int kk = 0; kk < 3; ++kk)
    for (int hh = 0; hh < 4; ++hh) {
      int ka = kk * 4 + k0, kb = ka + 1;
      w1b[kk][hh].x = (ka < 11) ? W1[ka * H + hh * 16 + nloc] : 0.f;
      w1b[kk][hh].y = (kb < 11) ? W1[kb * H + hh * 16 + nloc] : 0.f;
    }
  float b1v[4];
  for (int hh = 0; hh < 4; ++hh) b1v[hh] = b1[hh * 16 + nloc];
  float b2v[8];
  for (int oo = 0; oo < 8; ++oo) b2v[oo] = b2[oo * 16 + nloc];

  // Phase 1: gather msg_in rows into LDS (rows 41..47 duplicate self edge)
  const float* pi = pos + nbase * 3;
  for (int t = lane; t < KE * FIN; t += 32) {
    int e = t / FIN, f = t % FIN;
    int j = (e < K) ? idx_ws[nbase * K + e] : i;  // e==40: self loop
    float v = 0.f;
    if (f < F) v = x[((size_t)b * N + j) * F + f];
    else if (f < F + 3) v = pos[((size_t)b * N + j) * 3 + (f - F)] - pi[f - F];
    smsg[wid][e][f] = v;
  }
  __syncthreads();

  // Phase 2: h = relu(msg @ W1 + b1) -> LDS (C layout -> row-major)
  for (int r = 0; r < 3; ++r) {
    const int e = r * 16 + nloc;  // A: lane holds row e
    v2f a[3];
    for (int kk = 0; kk < 3; ++kk)  // ka = kk*4 + k0 is even -> aligned b64
      a[kk] = *(const v2f*)&smsg[wid][e][kk * 4 + k0];
    const int mbase = (lane < 16) ? 0 : 8;  // C: VGPR v -> row v (+8 hi half)
    for (int hh = 0; hh < 4; ++hh) {
      v8f c = {0.f, 0.f, 0.f, 0.f, 0.f, 0.f, 0.f, 0.f};
      for (int kk = 0; kk < 3; ++kk)
        c = __builtin_amdgcn_wmma_f32_16x16x4_f32(
            false, a[kk], false, w1b[kk][hh], (short)0, c, false, false);
      for (int v = 0; v < 8; ++v) {
        float hv = fmaxf(c[v] + b1v[hh], 0.f);
        sh[wid][r * 16 + mbase + v][hh * 16 + nloc] = hv;
      }
    }
  }
  __syncthreads();

  // Phase 3: m = h @ W2 (K=64 -> 16 WMMA steps), running max over edge rows.
  // Preload all A tiles (rows are WMMA-lane-major, pairs aligned), then for
  // each output tile stream 16 B operands from LDS through the WMMA chain.
  v2f a[3][16];
  for (int r = 0; r < 3; ++r)
    for (int kk = 0; kk < 16; ++kk)
      a[r][kk] = *(const v2f*)&sh[wid][r * 16 + nloc][kk * 4 + k0];

  float mx[8];
  for (int oo = 0; oo < 8; ++oo) {
    v2f bw[16];
    for (int kk = 0; kk < 16; ++kk)  // pair index p = (kk*4 + k0) / 2
      bw[kk] = sW2p[(kk * 2 + (k0 >> 1)) * O + oo * 16 + nloc];
    float t = -3.4e38f;
    for (int r = 0; r < 3; ++r) {
      v8f c = {0.f, 0.f, 0.f, 0.f, 0.f, 0.f, 0.f, 0.f};
      for (int kk = 0; kk < 16; ++kk)
        c = __builtin_amdgcn_wmma_f32_16x16x4_f32(
            false, a[r][kk], false, bw[kk], (short)0, c, false, false);
      for (int v = 0; v < 8; ++v) t = fmaxf(t, c[v]);
    }
    mx[oo] = t;
  }
  // merge lane halves: lanes 0-15 cover rows {v}, lanes 16-31 rows {v+8}
  for (int oo = 0; oo < 8; ++oo) {
    float other = __shfl_xor(mx[oo], 16, 32);
    mx[oo] = fmaxf(mx[oo], other);
  }
  if (lane < 16) {
    float* op = out + nbase * O;
    for (int oo = 0; oo < 8; ++oo) op[oo * 16 + nloc] = mx[oo] + b2v[oo];
  }
}

// --------------------------------------------------------------------------
extern "C" void kernel_launch(void* const* d_in, const int* in_sizes, int n_in,
                              void* d_out, int out_size, void* d_ws,
                              size_t ws_size, hipStream_t stream) {
  (void)in_sizes; (void)n_in; (void)out_size; (void)ws_size;
  const float* x  = (const float*)d_in[0];
  const float* pos = (const float*)d_in[1];
  const float* W1 = (const float*)d_in[2];
  const float* b1 = (const float*)d_in[3];
  const float* W2 = (const float*)d_in[4];
  const float* b2 = (const float*)d_in[5];
  float* out = (float*)d_out;
  int* idx_ws = (int*)d_ws;  // B*N*K ints = 5 MB

  // KNN: 8 batches x 16 tiles of 256 targets
  knn_topk_kernel<<<dim3(B * (N / 256)), dim3(256), 0, stream>>>(pos, idx_ws,
                                                                 out);
  // MLP + max aggregate: one wave per node, 2 waves per block
  pointconv_wmma_kernel<<<dim3((B * N) / WPB), dim3(32 * WPB), 0, stream>>>(
      x, pos, W1, b1, W2, b2, idx_ws, out);
}